// GNN_47648367182381
// MI455X (gfx1250) — compile-verified
//
#include <hip/hip_runtime.h>

// ---------------- problem constants (match reference) ----------------
#define NN   50000
#define EE   1600000
#define INCH 128
#define HIDF 64
#define NCLS 8
#define EPSN 1e-5f

typedef float v2f __attribute__((ext_vector_type(2)));
typedef float v8f __attribute__((ext_vector_type(8)));

// ---------------- utility ----------------
__global__ void gnn_zero_kernel(float* __restrict__ p, int n) {
  int i = blockIdx.x * blockDim.x + threadIdx.x;
  if (i < n) p[i] = 0.0f;
}

// deg[dst] += 1 over real edges
__global__ void gnn_deg_kernel(const int* __restrict__ dst, float* __restrict__ deg, int e) {
  int i = blockIdx.x * blockDim.x + threadIdx.x;
  if (i < e) atomicAdd(&deg[dst[i]], 1.0f);
}

// dinv = rsqrt(deg + 1)   (self-loop adds 1; deg_total >= 1 always)
__global__ void gnn_dinv_kernel(float* __restrict__ deg, int n) {
  int i = blockIdx.x * blockDim.x + threadIdx.x;
  if (i < n) deg[i] = rsqrtf(deg[i] + 1.0f);
}

// ---------------- fp32 WMMA GEMM: Y[nrows x 64] = A[nrows x K] @ W[K x 64] ----------------
// One wave32 computes one 16x16 tile of Y via V_WMMA_F32_16X16X4_F32.
// A-frag layout (ISA 7.12.2, 32-bit A 16x4): lane holds row M = lane%16,
//   K = 2*(lane/16) + {0,1} in its two regs. B symmetric (col N = lane%16).
// C/D: vgpr r, lane L -> M = r + 8*(L/16), N = L%16.
__global__ __launch_bounds__(256)
void gnn_gemm_wmma(const float* __restrict__ A, const float* __restrict__ W,
                   const float* __restrict__ bias, float* __restrict__ Y,
                   int nrows, int K, int do_relu) {
  const int lane  = threadIdx.x & 31;
  const int wv    = threadIdx.x >> 5;
  const int mbase = (blockIdx.x * 8 + wv) * 16;
  if (mbase >= nrows) return;                 // wave-uniform: EXEC stays all-1s for live waves
  const int nbase = blockIdx.y * 16;
  const int half  = lane >> 4;                // 0 or 1
  const int l16   = lane & 15;
  const int row   = mbase + l16;
  const int col   = nbase + l16;
  const int kk    = half * 2;

  v8f acc = {0.f, 0.f, 0.f, 0.f, 0.f, 0.f, 0.f, 0.f};
  const float* __restrict__ arow = A + (size_t)row * K;

  for (int k0 = 0; k0 < K; k0 += 4) {
    v2f a, b;
    a.x = arow[k0 + kk];
    a.y = arow[k0 + kk + 1];
    b.x = W[(size_t)(k0 + kk)     * HIDF + col];
    b.y = W[(size_t)(k0 + kk + 1) * HIDF + col];
    // (neg_a, A, neg_b, B, c_mod, C, reuse_a, reuse_b)
    acc = __builtin_amdgcn_wmma_f32_16x16x4_f32(false, a, false, b, (short)0, acc,
                                                false, false);
  }

  const float bv = bias ? bias[col] : 0.0f;
#pragma unroll
  for (int r = 0; r < 8; ++r) {
    int m = mbase + r + 8 * half;
    float v = acc[r] + bv;
    if (do_relu) v = fmaxf(v, 0.0f);
    Y[(size_t)m * HIDF + col] = v;
  }
}

// ---------------- edge scatter: out[dst] += dinv[src]*dinv[dst] * h[src] ----------------
// 64 threads per edge (one feature each); atomics resolve in L2 (h fits in L2).
__global__ __launch_bounds__(256)
void gnn_scatter_kernel(const float* __restrict__ h, const int* __restrict__ src,
                        const int* __restrict__ dst, const float* __restrict__ dinv,
                        float* __restrict__ out) {
  long long idx = (long long)blockIdx.x * blockDim.x + threadIdx.x;
  int e = (int)(idx >> 6);
  int f = (int)(idx & 63);
  if (e >= EE) return;
  int s = src[e], d = dst[e];
  float w = dinv[s] * dinv[d];
  atomicAdd(&out[(size_t)d * HIDF + f], w * h[(size_t)s * HIDF + f]);
}

// self-loop contribution: out[i] += dinv[i]^2 * h[i]
__global__ void gnn_selfloop_kernel(const float* __restrict__ h, const float* __restrict__ dinv,
                                    float* __restrict__ out, int n) {
  int idx = blockIdx.x * blockDim.x + threadIdx.x;
  if (idx >= n * HIDF) return;
  int i = idx >> 6;
  float di = dinv[i];
  out[idx] += di * di * h[idx];
}

// ---------------- BatchNorm (training mode, biased var) ----------------
// stats[f] = sum, stats[64+f] = sumsq, accumulated via block partials + atomics
__global__ __launch_bounds__(256)
void gnn_bn_stats_kernel(const float* __restrict__ x, float* __restrict__ stats, int n) {
  __shared__ float s_sum[256];
  __shared__ float s_sq[256];
  const int f   = threadIdx.x & 63;
  const int sub = threadIdx.x >> 6;   // 0..3
  const int base = blockIdx.x * 64;   // 64 rows per block
  float sum = 0.f, sq = 0.f;
#pragma unroll
  for (int i = 0; i < 16; ++i) {
    int r = base + sub + i * 4;
    if (r < n) {
      float v = x[(size_t)r * HIDF + f];
      sum += v;
      sq  += v * v;
    }
  }
  s_sum[threadIdx.x] = sum;
  s_sq[threadIdx.x]  = sq;
  __syncthreads();
  if (sub == 0) {
    sum = s_sum[f] + s_sum[64 + f] + s_sum[128 + f] + s_sum[192 + f];
    sq  = s_sq[f]  + s_sq[64 + f]  + s_sq[128 + f]  + s_sq[192 + f];
    atomicAdd(&stats[f], sum);
    atomicAdd(&stats[64 + f], sq);
  }
}

// ss[f] = scale, ss[64+f] = shift  (conv bias cancels under BN and is omitted)
__global__ void gnn_bn_final_kernel(const float* __restrict__ stats,
                                    const float* __restrict__ g, const float* __restrict__ b,
                                    float* __restrict__ ss, int n) {
  int f = threadIdx.x;
  if (f >= 64) return;
  float inv_n = 1.0f / (float)n;
  float mu  = stats[f] * inv_n;
  float var = stats[64 + f] * inv_n - mu * mu;
  float sc  = g[f] * rsqrtf(var + EPSN);
  ss[f]      = sc;
  ss[64 + f] = b[f] - mu * sc;
}

// x = relu(x*scale + shift), in place
__global__ void gnn_bn_apply_kernel(float* __restrict__ x, const float* __restrict__ ss, int total) {
  int idx = blockIdx.x * blockDim.x + threadIdx.x;
  if (idx >= total) return;
  int f = idx & 63;
  x[idx] = fmaxf(x[idx] * ss[f] + ss[64 + f], 0.0f);
}

// ---------------- LayerNorm over 64 features, one wave32 per row ----------------
__global__ __launch_bounds__(256)
void gnn_ln_kernel(float* __restrict__ x, const float* __restrict__ g,
                   const float* __restrict__ b, int n) {
  const int lane = threadIdx.x & 31;
  const int row  = blockIdx.x * 8 + (threadIdx.x >> 5);
  if (row >= n) return;
  float2 v = *(const float2*)&x[(size_t)row * HIDF + lane * 2];
  float sum = v.x + v.y;
  float sq  = v.x * v.x + v.y * v.y;
#pragma unroll
  for (int m = 16; m >= 1; m >>= 1) {
    sum += __shfl_xor(sum, m, 32);
    sq  += __shfl_xor(sq, m, 32);
  }
  float mu  = sum * (1.0f / 64.0f);
  float var = sq * (1.0f / 64.0f) - mu * mu;
  float inv = rsqrtf(var + EPSN);
  float2 o;
  o.x = (v.x - mu) * inv * g[lane * 2]     + b[lane * 2];
  o.y = (v.y - mu) * inv * g[lane * 2 + 1] + b[lane * 2 + 1];
  *(float2*)&x[(size_t)row * HIDF + lane * 2] = o;
}

// ---------------- final linear 64 -> 8 ----------------
__global__ void gnn_out_kernel(const float* __restrict__ x, const float* __restrict__ w,
                               const float* __restrict__ b, float* __restrict__ y, int n) {
  int idx = blockIdx.x * blockDim.x + threadIdx.x;
  if (idx >= n * NCLS) return;
  int row = idx >> 3, c = idx & 7;
  float s = b[c];
  const float* __restrict__ xr = x + (size_t)row * HIDF;
#pragma unroll
  for (int k = 0; k < HIDF; ++k) s += xr[k] * w[k * NCLS + c];
  y[idx] = s;
}

// =====================================================================
extern "C" void kernel_launch(void* const* d_in, const int* in_sizes, int n_in,
                              void* d_out, int out_size, void* d_ws, size_t ws_size,
                              hipStream_t stream) {
  // ---- inputs, flattened in setup_inputs() dict insertion order ----
  const float* x   = (const float*)d_in[0];                  // [N,128]
  const int*   ei  = (const int*)d_in[1];                    // [2,E]
  const int*   src = ei;
  const int*   dst = ei + EE;
  const float* convw[3] = {(const float*)d_in[2], (const float*)d_in[4], (const float*)d_in[6]};
  // conv biases d_in[3],[5],[7] are mathematically no-ops under BatchNorm -> skipped
  const float* bng[3] = {(const float*)d_in[8],  (const float*)d_in[10], (const float*)d_in[12]};
  const float* bnb[3] = {(const float*)d_in[9],  (const float*)d_in[11], (const float*)d_in[13]};
  const float *mlpw[4], *mlpb[4], *lng[4], *lnb[4];
  for (int l = 0; l < 4; ++l) {
    mlpw[l] = (const float*)d_in[14 + 4 * l];
    mlpb[l] = (const float*)d_in[15 + 4 * l];
    lng[l]  = (const float*)d_in[16 + 4 * l];
    lnb[l]  = (const float*)d_in[17 + 4 * l];
  }
  const float* outw = (const float*)d_in[30];                // [64,8]
  const float* outb = (const float*)d_in[31];                // [8]

  // ---- workspace carve (~26 MB) ----
  float* bufH   = (float*)d_ws;                 // [N,64] GEMM output h
  float* bufAgg = bufH + (size_t)NN * HIDF;     // [N,64] aggregate / activation
  float* dinv   = bufAgg + (size_t)NN * HIDF;   // [N]    degree -> dinv
  float* stats  = dinv + NN;                    // [128]  BN sum/sumsq
  float* ss     = stats + 128;                  // [128]  BN scale/shift

  // ---- degrees (once) ----
  gnn_zero_kernel<<<(NN + 255) / 256, 256, 0, stream>>>(dinv, NN);
  gnn_deg_kernel<<<(EE + 255) / 256, 256, 0, stream>>>(dst, dinv, EE);
  gnn_dinv_kernel<<<(NN + 255) / 256, 256, 0, stream>>>(dinv, NN);

  const int elem = NN * HIDF;
  const dim3 gemm_grid((NN / 16 + 7) / 8, HIDF / 16);        // 391 x 4 waves-of-tiles

  // ---- GCN layers ----
  const float* cur = x;
  int K = INCH;
  for (int l = 0; l < 3; ++l) {
    gnn_gemm_wmma<<<gemm_grid, 256, 0, stream>>>(cur, convw[l], nullptr, bufH, NN, K, 0);
    gnn_zero_kernel<<<(elem + 255) / 256, 256, 0, stream>>>(bufAgg, elem);
    gnn_scatter_kernel<<<(EE * HIDF) / 256, 256, 0, stream>>>(bufH, src, dst, dinv, bufAgg);
    gnn_selfloop_kernel<<<(elem + 255) / 256, 256, 0, stream>>>(bufH, dinv, bufAgg, NN);
    gnn_zero_kernel<<<1, 256, 0, stream>>>(stats, 128);
    gnn_bn_stats_kernel<<<(NN + 63) / 64, 256, 0, stream>>>(bufAgg, stats, NN);
    gnn_bn_final_kernel<<<1, 64, 0, stream>>>(stats, bng[l], bnb[l], ss, NN);
    gnn_bn_apply_kernel<<<(elem + 255) / 256, 256, 0, stream>>>(bufAgg, ss, elem);
    cur = bufAgg;
    K = HIDF;
  }

  // ---- MLP layers (Linear+bias+ReLU fused in GEMM epilogue, then LayerNorm) ----
  float* a = bufAgg;
  float* o = bufH;
  for (int l = 0; l < 4; ++l) {
    gnn_gemm_wmma<<<gemm_grid, 256, 0, stream>>>(a, mlpw[l], mlpb[l], o, NN, HIDF, 1);
    gnn_ln_kernel<<<(NN + 7) / 8, 256, 0, stream>>>(o, lng[l], lnb[l], NN);
    float* t = a; a = o; o = t;
  }

  // ---- output head ----
  gnn_out_kernel<<<(NN * NCLS + 255) / 256, 256, 0, stream>>>(a, outw, outb, (float*)d_out, NN);
}